// GraphConvolution_35545149342388
// MI455X (gfx1250) — compile-verified
//
#include <hip/hip_runtime.h>

#define IN_DIM 256
#define OUT_DIM 128
#define KCHUNK 64
#define NCHUNKS (IN_DIM / KCHUNK) // 4
#define XSTRIDE 68                // floats per X row in LDS (bank-conflict-free)
#define XTILE (128 * XSTRIDE)     // one X buffer: 128 rows x 64 K (padded)
#define WSTRIDE 288               // floats per K-pair row of swizzled W (mod 64 == 32)
#define WTOT ((IN_DIM / 2) * WSTRIDE)
#define SMEM_FLOATS (2 * XTILE + WTOT) // 54272 floats = 217,088 B dynamic LDS

typedef __attribute__((ext_vector_type(2))) float v2f;
typedef __attribute__((ext_vector_type(8))) float v8f;

// Async-to-LDS builtin signature (confirmed by clang diagnostic):
//   (v4i addrspace(1)*, v4i addrspace(3)*, Ii offset, Ii cpol)
typedef __attribute__((__vector_size__(4 * sizeof(int)))) int v4i;
typedef __attribute__((address_space(1))) v4i* gptr_v4i;
typedef __attribute__((address_space(3))) v4i* lptr_v4i;

__device__ static inline void wait_async_le8() {
#if __has_builtin(__builtin_amdgcn_s_wait_asynccnt)
  __builtin_amdgcn_s_wait_asynccnt(8);
#else
  asm volatile("s_wait_asynccnt 8" ::: "memory");
#endif
}
__device__ static inline void wait_async_le0() {
#if __has_builtin(__builtin_amdgcn_s_wait_asynccnt)
  __builtin_amdgcn_s_wait_asynccnt(0);
#else
  asm volatile("s_wait_asynccnt 0" ::: "memory");
#endif
}

// Issue 8 async b128 copies per thread: X[mblock..mblock+127, kb..kb+63] -> xb
__device__ static inline void stage_x_async(const float* __restrict__ X,
                                            float* xb, int mblock, int n_rows,
                                            int kb, int tid) {
#pragma unroll
  for (int j = 0; j < 8; ++j) {
    const int q = tid + j * 256; // 0..2047 quads
    const int row = q >> 4;      // 0..127
    const int qc = q & 15;       // 0..15 quad-of-16B within the 64-float chunk
    int mr = mblock + row;
    if (mr >= n_rows) mr = n_rows - 1; // clamped duplicate read, never OOB
    const float* src = X + (long)mr * IN_DIM + kb + qc * 4;
    float* dst = xb + row * XSTRIDE + qc * 4;
    __builtin_amdgcn_global_load_async_to_lds_b128(
        (gptr_v4i)(unsigned long long)src,
        (lptr_v4i)(unsigned int)(unsigned long long)dst, 0, 0);
  }
}

// -------- Phase 1: support = X @ W via fp32 WMMA (16x16x4) ------------------
// Block = 256 threads = 8 waves; 128-row x 128-col tile per block.
// X tiles double-buffered through the async-to-LDS engine; W staged once,
// swizzled so each B fragment is a single aligned ds_load_b64.
__global__ __launch_bounds__(256) void gcn_gemm_wmma(
    const float* __restrict__ X, const float* __restrict__ W,
    float* __restrict__ support, int n_rows) {
  extern __shared__ float smem[];
  float* Wl2 = smem + 2 * XTILE;

  const int tid = threadIdx.x;
  const int wv = tid >> 5;
  const int lane = tid & 31;
  const int hf = lane >> 4;  // K-half selector (0/1)
  const int l15 = lane & 15; // M or N within tile
  const int mblock = blockIdx.x * 128;
  const int mbase = mblock + wv * 16;

  // Pipeline prologue: async-stage X chunk 0, swizzle-stage all of W,
  // async-stage X chunk 1, then retire chunk 0.
  stage_x_async(X, smem, mblock, n_rows, 0, tid);

  for (int i = tid; i < (IN_DIM * OUT_DIM) / 4; i += 256) { // 32 iters
    const int r = (i * 4) / OUT_DIM;
    const int c = (i * 4) % OUT_DIM;
    const float4 f = *(const float4*)(W + (long)r * OUT_DIM + c);
    float* dst = Wl2 + (r >> 1) * WSTRIDE + 2 * c + (r & 1);
    dst[0] = f.x;
    dst[2] = f.y;
    dst[4] = f.z;
    dst[6] = f.w;
  }

  stage_x_async(X, smem + XTILE, mblock, n_rows, KCHUNK, tid);
  wait_async_le8(); // own chunk-0 ops retired (in-order)
  __syncthreads();  // everyone's chunk 0 + W visible

  v8f acc[8] = {};

  for (int kb4 = 0; kb4 < NCHUNKS; ++kb4) {
    const float* xb = smem + (kb4 & 1) * XTILE;
    // A: this lane covers X[mblock+l15, kb + 4c + 2hf + {0,1}]
    const float* abase = xb + l15 * XSTRIDE + 2 * hf;
    // B: pair index (kb/2 + 2c + hf), column 2*n
    const float* wbase = Wl2 + (kb4 * (KCHUNK / 2) + hf) * WSTRIDE + 2 * l15;

#pragma unroll
    for (int c = 0; c < KCHUNK / 4; ++c) {
      const float2 af = *(const float2*)(abase + 4 * c);
      v2f a;
      a.x = af.x;
      a.y = af.y;
      const float* wc = wbase + 2 * c * WSTRIDE;
#pragma unroll
      for (int nt = 0; nt < 8; ++nt) {
        const float2 bf = *(const float2*)(wc + 32 * nt);
        v2f b;
        b.x = bf.x;
        b.y = bf.y;
        acc[nt] = __builtin_amdgcn_wmma_f32_16x16x4_f32(
            false, a, false, b, (short)0, acc[nt], false, false);
      }
    }

    __syncthreads(); // all waves done reading xb before overwrite
    if (kb4 + 2 < NCHUNKS) {
      stage_x_async(X, smem + (kb4 & 1) * XTILE, mblock, n_rows,
                    (kb4 + 2) * KCHUNK, tid);
      wait_async_le8(); // oldest 8 (chunk kb4+1) retired; newest 8 in flight
    } else {
      wait_async_le0(); // drain (chunk kb4+1, if any)
    }
    __syncthreads(); // publish chunk kb4+1 to all waves
  }

  // D layout: VGPR v -> row mbase + 8*hf + v, col nt*16 + l15
#pragma unroll
  for (int nt = 0; nt < 8; ++nt) {
#pragma unroll
    for (int v = 0; v < 8; ++v) {
      const int mr = mbase + 8 * hf + v;
      if (mr < n_rows)
        support[(long)mr * OUT_DIM + nt * 16 + l15] = acc[nt][v];
    }
  }
}

// -------- Phase 2: out[row] += val * support[col]  (edge-parallel) ----------
// One wave per edge: 32 lanes x float4 = 128 floats, coalesced 512B segments.
__global__ __launch_bounds__(256) void gcn_spmm_scatter(
    const float* __restrict__ support, const int* __restrict__ erow,
    const int* __restrict__ ecol, const float* __restrict__ eval,
    float* __restrict__ out, int n_edges) {
  const int lane = threadIdx.x & 31;
  const int wave = (int)((blockIdx.x * blockDim.x + threadIdx.x) >> 5);
  const int n_waves = (int)((gridDim.x * blockDim.x) >> 5);

  for (int e = wave; e < n_edges; e += n_waves) {
    const int r = erow[e];
    const int c = ecol[e];
    const float v = eval[e];
    const float4 g = *(const float4*)(support + (long)c * OUT_DIM + lane * 4);
    float* dst = out + (long)r * OUT_DIM + lane * 4;
    atomicAdd(dst + 0, g.x * v);
    atomicAdd(dst + 1, g.y * v);
    atomicAdd(dst + 2, g.z * v);
    atomicAdd(dst + 3, g.w * v);
  }
}

// -------- Zero-init of the output accumulator -------------------------------
__global__ void gcn_zero(float* __restrict__ p, int n) {
  const int i4 = (int)(blockIdx.x * blockDim.x + threadIdx.x) * 4;
  if (i4 + 3 < n) {
    *(float4*)(p + i4) = make_float4(0.f, 0.f, 0.f, 0.f);
  } else {
    for (int j = i4; j < n; ++j) p[j] = 0.f;
  }
}

extern "C" void kernel_launch(void* const* d_in, const int* in_sizes, int n_in,
                              void* d_out, int out_size, void* d_ws,
                              size_t ws_size, hipStream_t stream) {
  const float* x = (const float*)d_in[0];     // [N, 256]
  const float* w = (const float*)d_in[1];     // [256, 128]
  const int* erow = (const int*)d_in[2];      // [E]
  const int* ecol = (const int*)d_in[3];      // [E]
  const float* eval = (const float*)d_in[4];  // [E]
  float* out = (float*)d_out;                 // [N, 128]
  float* support = (float*)d_ws;              // scratch: [N, 128] fp32

  const int n_nodes = in_sizes[0] / IN_DIM;
  const int n_edges = in_sizes[2];
  const int out_elems = n_nodes * OUT_DIM;

  // 1) zero the accumulator output
  {
    const int threads = 256;
    const int blocks = (out_elems / 4 + threads - 1) / threads;
    gcn_zero<<<blocks, threads, 0, stream>>>(out, out_elems);
  }
  // 2) support = X @ W (WMMA, async-pipelined tiles)
  {
    const int blocks = (n_nodes + 127) / 128;
    const size_t shbytes = (size_t)SMEM_FLOATS * sizeof(float); // 217,088 B
    gcn_gemm_wmma<<<blocks, 256, shbytes, stream>>>(x, w, support, n_nodes);
  }
  // 3) scatter-accumulate over edges
  {
    gcn_spmm_scatter<<<2048, 256, 0, stream>>>(support, erow, ecol, eval, out,
                                               n_edges);
  }
}